// NormalAugmenter_53171695124702
// MI455X (gfx1250) — compile-verified
//
#include <hip/hip_runtime.h>
#include <stdint.h>

#define AS1 __attribute__((address_space(1)))
#define AS3 __attribute__((address_space(3)))

typedef unsigned int u32x4 __attribute__((ext_vector_type(4)));
typedef int i32x4 __attribute__((ext_vector_type(4)));
typedef int i32x8 __attribute__((ext_vector_type(8)));

static constexpr int D_DIM = 1024;       // feature dim (per reference)
static constexpr int RPB = 8;            // rows per block, one wave32 per row
static constexpr int THREADS = RPB * 32; // 256 threads = 8 wave32
static constexpr float IN_EPS = 1e-5f;

// ---------------------------------------------------------------------------
// Cheap deterministic Gaussian pair per element:
// one 32-bit avalanche hash -> two 16-bit uniforms -> Box-Muller with raw
// v_log_f32 / v_sqrt_f32 / v_sin_f32 / v_cos_f32 (no libm range reduction).
// ~20 VALU + 4 TRANS per element, tails clipped at 4.7 sigma.
// ---------------------------------------------------------------------------
__device__ __forceinline__ float augment_one(float x, uint32_t idx, float rstd,
                                             float nmr, float hs, float hm) {
  uint32_t h = idx;
  h ^= h >> 16; h *= 0x21F0AAADu;
  h ^= h >> 15; h *= 0x735A2D97u;
  h ^= h >> 15;
  // u1 in (0,1]: 16-bit, forced nonzero. log2 -> ln folded into one multiply.
  float u1 = (float)((h >> 16) | 1u) * 1.52587890625e-05f;       // /65536
  float l2 = __builtin_amdgcn_logf(u1);                          // log2(u1)
  float rmag = __builtin_amdgcn_sqrtf(l2 * -1.3862943611198906f);// sqrt(-2 ln u1)
  float th = (float)(h & 0xFFFFu) * 9.58737992428526e-05f;       // 2*pi/65536
  float ea = rmag * __builtin_amdgcn_cosf(th);                   // N(0,1)
  float eb = rmag * __builtin_amdgcn_sinf(th);                   // N(0,1)
  // out = (0.5 + hs*ea) * n + hm + hs*eb,  n = (x-mean)*rstd
  float n = __builtin_fmaf(x, rstd, nmr);
  float t = __builtin_fmaf(ea, n, eb);
  return __builtin_fmaf(hs, t, __builtin_fmaf(0.5f, n, hm));
}

// ---------- wave32 butterfly sum via ds_swizzle (xor 1,2,4,8,16) ----------
__device__ __forceinline__ float wave_reduce_add(float v) {
  v += __int_as_float(__builtin_amdgcn_ds_swizzle(__float_as_int(v), 0x041F));
  v += __int_as_float(__builtin_amdgcn_ds_swizzle(__float_as_int(v), 0x081F));
  v += __int_as_float(__builtin_amdgcn_ds_swizzle(__float_as_int(v), 0x101F));
  v += __int_as_float(__builtin_amdgcn_ds_swizzle(__float_as_int(v), 0x201F));
  v += __int_as_float(__builtin_amdgcn_ds_swizzle(__float_as_int(v), 0x401F));
  return v;
}

__global__ __launch_bounds__(THREADS) void
normal_augmenter_kernel(const float* __restrict__ in, float* __restrict__ out,
                        int rows) {
  __shared__ float smem[RPB * D_DIM]; // 32 KB tile (<< 320 KB/WGP)

  const int lane = threadIdx.x & 31;
  const int wv = threadIdx.x >> 5;
  const int row0 = (int)blockIdx.x * RPB;

  // ================= stage global -> LDS =================
#if __has_builtin(__builtin_amdgcn_tensor_load_to_lds)
  if (threadIdx.x < 32) { // wave 0 issues one TDM transfer for the whole block
    AS3 float* sm3 = (AS3 float*)smem;
    uint32_t lds_base = (uint32_t)(uintptr_t)sm3;
    uint64_t ga = (uint64_t)(uintptr_t)(in + (int64_t)row0 * D_DIM);
    int nri = rows - row0;
    uint32_t nr = (uint32_t)(nri > RPB ? RPB : nri);

    // D# group 0: count=1 | lds_addr | global_addr[56:0] | type=2
    u32x4 g0;
    g0[0] = 1u;
    g0[1] = lds_base;
    g0[2] = (uint32_t)ga;
    g0[3] = ((uint32_t)(ga >> 32) & 0x01FFFFFFu) | (2u << 30);

    // D# group 1: data_size=4B; tensor 1024 x nr, stride0=1024; tile 1024 x nr
    i32x8 g1;
    g1[0] = (2 << 16);                                        // data_size = 4B
    g1[1] = (int)(((uint32_t)D_DIM & 0xFFFFu) << 16);         // tensor_dim0 lo16
    g1[2] = (int)(((uint32_t)D_DIM >> 16) | ((nr & 0xFFFFu) << 16));
    g1[3] = (int)((nr >> 16) | (((uint32_t)D_DIM & 0xFFFFu) << 16));
    g1[4] = (int)nr;                                          // tile_dim1|tile_dim2=0
    g1[5] = D_DIM;                                            // stride0 lo32
    g1[6] = 0;
    g1[7] = 0;

    i32x4 z4 = {0, 0, 0, 0}; // groups 2/3 unused (2D tensor)
#if __clang_major__ >= 23
    i32x8 z8 = {0, 0, 0, 0, 0, 0, 0, 0};
    __builtin_amdgcn_tensor_load_to_lds(g0, g1, z4, z4, z8, 0);
#else
    __builtin_amdgcn_tensor_load_to_lds(g0, g1, z4, z4, 0);
#endif
#if __has_builtin(__builtin_amdgcn_s_wait_tensorcnt)
    __builtin_amdgcn_s_wait_tensorcnt(0);
#else
    asm volatile("s_wait_tensorcnt 0x0" ::: "memory");
#endif
  }
#else
  // Fallback: each wave copies its own row (async-to-LDS if available)
  {
    const int frow = row0 + wv;
    if (frow < rows) {
      const float* grow = in + (int64_t)frow * D_DIM;
#if __has_builtin(__builtin_amdgcn_global_load_async_to_lds_b128)
      AS3 char* lrow = (AS3 char*)((AS3 float*)smem + wv * D_DIM);
#pragma unroll
      for (int j = 0; j < D_DIM / 128; ++j) {
        const int off = (j * 32 + lane) * 16;
        __builtin_amdgcn_global_load_async_to_lds_b128(
            (AS1 const void*)((const char*)grow + off),
            (AS3 void*)(lrow + off), 0, 0);
      }
#if __has_builtin(__builtin_amdgcn_s_wait_asynccnt)
      __builtin_amdgcn_s_wait_asynccnt(0);
#else
      asm volatile("s_wait_asynccnt 0x0" ::: "memory");
#endif
#else
      float4* srow = (float4*)(smem + wv * D_DIM);
      const float4* g4 = (const float4*)grow;
#pragma unroll
      for (int j = 0; j < D_DIM / 128; ++j)
        srow[j * 32 + lane] = g4[j * 32 + lane];
#endif
    }
  }
#endif
  __syncthreads();

  // ================= per-row compute: one wave32 per 1024-elem row =========
  const int row = row0 + wv;
  if (row >= rows) return;

  const float4* lrow4 = (const float4*)(smem + wv * D_DIM); // ds_load_b128
  float4 vals[D_DIM / 128]; // 32 floats/lane resident for the apply pass
  float s = 0.f, ss = 0.f;
#pragma unroll
  for (int j = 0; j < D_DIM / 128; ++j) {
    float4 v = lrow4[j * 32 + lane];
    vals[j] = v;
    s += (v.x + v.y) + (v.z + v.w);
    ss += (v.x * v.x + v.y * v.y) + (v.z * v.z + v.w * v.w);
  }
  s = wave_reduce_add(s);
  ss = wave_reduce_add(ss);

  const float inv_d = 1.0f / (float)D_DIM;
  float mean = s * inv_d;
  float var_b = ss * inv_d - mean * mean;                    // biased (InstanceNorm)
  var_b = var_b < 0.f ? 0.f : var_b;
  float std_u = __builtin_amdgcn_sqrtf(var_b * ((float)D_DIM / (float)(D_DIM - 1)));
  float rstd = __builtin_amdgcn_rsqf(var_b + IN_EPS);

  // hoisted per-row constants for the 4-op FMA apply chain
  const float hs = 0.5f * std_u;      // half unbiased std
  const float hm = 0.5f * mean;       // half mean
  const float nmr = -mean * rstd;     // n = fma(x, rstd, nmr)

  float4* orow = (float4*)(out + (int64_t)row * D_DIM);
  const uint32_t base_idx = (uint32_t)row * (uint32_t)D_DIM; // < 2^26, fits u32
#pragma unroll
  for (int j = 0; j < D_DIM / 128; ++j) {
    const int c = j * 32 + lane; // float4 slot within the row
    float4 v = vals[j];
    uint32_t e = base_idx + (uint32_t)c * 4u;
    float4 o;
    o.x = augment_one(v.x, e + 0u, rstd, nmr, hs, hm);
    o.y = augment_one(v.y, e + 1u, rstd, nmr, hs, hm);
    o.z = augment_one(v.z, e + 2u, rstd, nmr, hs, hm);
    o.w = augment_one(v.w, e + 3u, rstd, nmr, hs, hm);
    orow[c] = o; // global_store_b128
  }
}

extern "C" void kernel_launch(void* const* d_in, const int* in_sizes, int n_in,
                              void* d_out, int out_size, void* d_ws,
                              size_t ws_size, hipStream_t stream) {
  (void)n_in; (void)out_size; (void)d_ws; (void)ws_size;
  const float* feat = (const float*)d_in[0];
  float* out = (float*)d_out;
  const int rows = in_sizes[0] / D_DIM; // 128*256 = 32768
  const int blocks = (rows + RPB - 1) / RPB;
  normal_augmenter_kernel<<<blocks, THREADS, 0, stream>>>(feat, out, rows);
}